// PPI_predictor_41016937676830
// MI455X (gfx1250) — compile-verified
//
#include <hip/hip_runtime.h>
#include <hip/hip_bf16.h>

typedef __attribute__((ext_vector_type(2))) float v2f;
typedef __attribute__((ext_vector_type(8))) float v8f;

#define DIM      10
#define PADDIM   12
#define KS       23
#define KTOP     55
#define ROWS_BLK 128
#define HALO     11
#define TILE_ROWS (ROWS_BLK + 2 * HALO)   // 150
#define CHUNK    8192
#define MAXCH    128                       // max chunks (L up to ~1M)

// ---------------------------------------------------------------------------
// Sortable-key mapping for float top-k (monotonic: bigger float -> bigger key)
// ---------------------------------------------------------------------------
__device__ __forceinline__ unsigned f2key(float f) {
    unsigned u = __float_as_uint(f);
    return (u & 0x80000000u) ? ~u : (u | 0x80000000u);
}
__device__ __forceinline__ float key2f(unsigned k) {
    unsigned u = (k & 0x80000000u) ? (k ^ 0x80000000u) : ~k;
    return __uint_as_float(u);
}

// ---------------------------------------------------------------------------
// Prep: build WMMA B-fragments for the 3 conv layers.
// Conv layer == sum over kh of  x[i+kh-11,:] @ M_kh,  M_kh[c][j] = w[kh][c-j+11]
// B fragment (4x16, f32): lanes 0-15 hold rows {0,1} in vgprs {0,1},
//                         lanes 16-31 hold rows {2,3}.
// wfrag[((l*23+kh)*3+g)*64 + lane*2 + t]
// ---------------------------------------------------------------------------
__global__ void prep_wfrag_kernel(const float* __restrict__ w_conv,
                                  float* __restrict__ wfrag) {
    const int TOT = 3 * KS * 3 * 32 * 2;
    for (int idx = threadIdx.x; idx < TOT; idx += blockDim.x) {
        int t    = idx & 1;
        int lane = (idx >> 1) & 31;
        int cg   = idx >> 6;
        int g    = cg % 3;
        int kh   = (cg / 3) % KS;
        int l    = cg / (3 * KS);
        int j    = lane & 15;
        int krow = ((lane >> 4) << 1) + t;  // 0..3
        int ch   = g * 4 + krow;            // input channel
        float v = 0.0f;
        if (ch < DIM && j < DIM) {
            int kw = ch - j + HALO;
            if (kw >= 0 && kw < KS) v = w_conv[(l * KS + kh) * KS + kw];
        }
        wfrag[idx] = v;
    }
}

// ---------------------------------------------------------------------------
// Conv layer with V_WMMA_F32_16X16X4_F32.
// Block = 256 threads = 8 waves; each wave computes a 16-row x 10-col tile.
// Layer-1 fuses the embedding gather (gather != 0).
// ---------------------------------------------------------------------------
__global__ __launch_bounds__(256) void conv_wmma_kernel(
    const float* __restrict__ in, const int* __restrict__ pidx,
    const float* __restrict__ embed, const float* __restrict__ wfrag_l,
    const float* __restrict__ bconv, int layer,
    float* __restrict__ out, int L, int gather)
{
    __shared__ float xt[TILE_ROWS][PADDIM];
    __shared__ float bf[KS * 3 * 64];

    const int tid  = threadIdx.x;
    const int row0 = blockIdx.x * ROWS_BLK;

    // B fragments -> LDS
    for (int i = tid; i < KS * 3 * 64; i += 256) bf[i] = wfrag_l[i];

    // Input tile (with halo) -> LDS; rows outside [0,L) are zero (conv padding)
    for (int i = tid; i < TILE_ROWS * DIM; i += 256) {
        int r = i / DIM, c = i % DIM;
        int gr = row0 - HALO + r;
        float v = 0.0f;
        if (gr >= 0 && gr < L)
            v = gather ? embed[pidx[gr] * DIM + c] : in[(long)gr * DIM + c];
        xt[r][c] = v;
    }
    // zero the two pad columns (channels 10,11)
    for (int i = tid; i < TILE_ROWS * 2; i += 256) xt[i >> 1][DIM + (i & 1)] = 0.0f;
    __syncthreads();

    const int wave = tid >> 5;        // 0..7 (wave32)
    const int lane = tid & 31;
    const int lh   = lane & 15;
    const int hi   = lane >> 4;
    const int wrow = wave * 16;       // local row base of this wave's tile

    v8f acc = {0.f, 0.f, 0.f, 0.f, 0.f, 0.f, 0.f, 0.f};

    for (int kh = 0; kh < KS; ++kh) {
        const int r = wrow + lh + kh;               // LDS row for A fragment
        #pragma unroll
        for (int g = 0; g < 3; ++g) {
            // A 16x4: lanes 0-15 -> K={0,1}; lanes 16-31 -> K={2,3}
            const int c0 = g * 4 + hi * 2;
            v2f a; a.x = xt[r][c0]; a.y = xt[r][c0 + 1];
            const float* bp = &bf[(kh * 3 + g) * 64 + lane * 2];
            v2f b; b.x = bp[0]; b.y = bp[1];
            acc = __builtin_amdgcn_wmma_f32_16x16x4_f32(
                false, a, false, b, (short)0, acc, false, false);
        }
    }

    // C layout: vgpr v, lanes 0-15 -> (M=v,   N=lane)
    //                   lanes16-31 -> (M=v+8, N=lane-16)
    const float bias = bconv[layer];
    if (lh < DIM) {
        #pragma unroll
        for (int v = 0; v < 8; ++v) {
            int grow = row0 + wrow + v + hi * 8;
            if (grow < L) {
                float o = acc[v] + bias;
                out[(long)grow * DIM + lh] = o > 0.0f ? o : 0.0f;
            }
        }
    }
}

// ---------------------------------------------------------------------------
// Top-k stage A: per (chunk, column), radix-threshold select 55 candidates.
// ---------------------------------------------------------------------------
__global__ __launch_bounds__(256) void topk_stageA_kernel(
    const float* __restrict__ in, unsigned* __restrict__ cand, int L, int nCh)
{
    __shared__ unsigned keys[CHUNK];
    __shared__ unsigned s_cnt;

    const int tid = threadIdx.x;
    const int col = blockIdx.y;
    const long base = (long)blockIdx.x * CHUNK;

    for (int i = tid; i < CHUNK; i += 256) {
        long row = base + i;
        keys[i] = (row < L) ? f2key(in[row * DIM + col]) : 0u;
    }
    __syncthreads();

    // largest T with count(key >= T) >= KTOP
    unsigned lo = 0u, hi = 0xFFFFFFFFu;
    while (lo < hi) {
        unsigned mid = (unsigned)(((unsigned long long)lo + hi + 1ull) >> 1);
        if (tid == 0) s_cnt = 0;
        __syncthreads();
        unsigned c = 0;
        for (int i = tid; i < CHUNK; i += 256) c += (keys[i] >= mid);
        atomicAdd(&s_cnt, c);
        __syncthreads();
        unsigned total = s_cnt;
        __syncthreads();
        if (total >= (unsigned)KTOP) lo = mid; else hi = mid - 1u;
    }
    const unsigned T = lo;

    if (tid == 0) s_cnt = 0;
    __syncthreads();
    unsigned* outp = cand + ((long)col * nCh + blockIdx.x) * KTOP;
    for (int i = tid; i < CHUNK; i += 256) {
        if (keys[i] > T) {
            unsigned pos = atomicAdd(&s_cnt, 1u);
            if (pos < (unsigned)KTOP) outp[pos] = keys[i];
        }
    }
    __syncthreads();
    unsigned ngt = s_cnt;
    if (ngt > (unsigned)KTOP) ngt = KTOP;
    for (unsigned i = ngt + tid; i < (unsigned)KTOP; i += 256) outp[i] = T;
}

// ---------------------------------------------------------------------------
// Top-k stage B: per column, rank candidates -> sorted top-55 ([KTOP, DIM]).
// ---------------------------------------------------------------------------
__global__ __launch_bounds__(256) void topk_stageB_kernel(
    const unsigned* __restrict__ cand, float* __restrict__ t_out, int nCh)
{
    __shared__ unsigned keys[MAXCH * KTOP];
    const int tid = threadIdx.x;
    const int col = blockIdx.x;
    const int N = nCh * KTOP;

    for (int i = tid; i < N; i += 256) keys[i] = cand[(long)col * N + i];
    __syncthreads();

    for (int i = tid; i < N; i += 256) {
        unsigned ki = keys[i];
        int rank = 0;
        for (int j = 0; j < N; ++j) {
            unsigned kj = keys[j];
            rank += (kj > ki) || (kj == ki && j < i);
        }
        if (rank < KTOP) t_out[rank * DIM + col] = key2f(ki);
    }
}

// ---------------------------------------------------------------------------
// Head: ff linears, 2-head attention (shared attn), output proj, concat, [2].
// ---------------------------------------------------------------------------
__global__ __launch_bounds__(256) void head_kernel(
    const float* __restrict__ t1in, const float* __restrict__ t2in,
    const float* __restrict__ Wff1, const float* __restrict__ bff1,
    const float* __restrict__ Wff2, const float* __restrict__ bff2,
    const float* __restrict__ Wl,   const float* __restrict__ bl,
    const float* __restrict__ Wint, const float* __restrict__ bint,
    float* __restrict__ out)
{
    __shared__ float a1[KTOP * DIM], a2[KTOP * DIM];
    __shared__ float q[KTOP * DIM], kk[KTOP * DIM];
    __shared__ float v1[KTOP * DIM], v2[KTOP * DIM];
    __shared__ float attn[2 * KTOP * KTOP];
    __shared__ float r1[DIM], r2[DIM];
    const int tid = threadIdx.x;

    // t @ Wff^T + bff
    for (int i = tid; i < KTOP * DIM; i += 256) {
        int k = i / DIM, d = i % DIM;
        float s1 = bff1[d], s2 = bff2[d];
        for (int c = 0; c < DIM; ++c) {
            s1 += t1in[k * DIM + c] * Wff1[d * DIM + c];
            s2 += t2in[k * DIM + c] * Wff2[d * DIM + c];
        }
        a1[i] = s1; a2[i] = s2;
    }
    __syncthreads();

    // q (a1,Wl0) / k (a2,Wl1) / val1 (a1,Wl2) / val2 (a2,Wl3)
    for (int i = tid; i < KTOP * DIM; i += 256) {
        int k = i / DIM, d = i % DIM;
        float sq = bl[0 * DIM + d], sk = bl[1 * DIM + d];
        float s1 = bl[2 * DIM + d], s2 = bl[3 * DIM + d];
        for (int c = 0; c < DIM; ++c) {
            float x1 = a1[k * DIM + c], x2 = a2[k * DIM + c];
            sq += x1 * Wl[(0 * DIM + d) * DIM + c];
            sk += x2 * Wl[(1 * DIM + d) * DIM + c];
            s1 += x1 * Wl[(2 * DIM + d) * DIM + c];
            s2 += x2 * Wl[(3 * DIM + d) * DIM + c];
        }
        q[i] = sq; kk[i] = sk; v1[i] = s1; v2[i] = s2;
    }
    __syncthreads();

    // scores[h][qi][kj] = q_h(qi) . k_h(kj)   (D_K = 5)
    for (int i = tid; i < 2 * KTOP * KTOP; i += 256) {
        int h = i / (KTOP * KTOP);
        int rem = i % (KTOP * KTOP);
        int qi = rem / KTOP, kj = rem % KTOP;
        float s = 0.0f;
        for (int d = 0; d < 5; ++d)
            s += q[qi * DIM + h * 5 + d] * kk[kj * DIM + h * 5 + d];
        attn[i] = s;
    }
    __syncthreads();

    // softmax over last dim (110 rows)
    for (int r = tid; r < 2 * KTOP; r += 256) {
        float* row = &attn[r * KTOP];
        float m = row[0];
        for (int j = 1; j < KTOP; ++j) m = fmaxf(m, row[j]);
        float s = 0.0f;
        for (int j = 0; j < KTOP; ++j) { float e = expf(row[j] - m); row[j] = e; s += e; }
        float inv = 1.0f / s;
        for (int j = 0; j < KTOP; ++j) row[j] *= inv;
    }
    __syncthreads();

    // x1 = attn @ v1, x2 = attn @ v2  (reuse a1/a2 as x storage, [K, DIM])
    for (int i = tid; i < 2 * KTOP * 5; i += 256) {
        int h = i / (KTOP * 5);
        int rem = i % (KTOP * 5);
        int qi = rem / 5, d = rem % 5;
        const float* arow = &attn[(h * KTOP + qi) * KTOP];
        float s1 = 0.0f, s2 = 0.0f;
        for (int j = 0; j < KTOP; ++j) {
            float a = arow[j];
            s1 += a * v1[j * DIM + h * 5 + d];
            s2 += a * v2[j * DIM + h * 5 + d];
        }
        a1[qi * DIM + h * 5 + d] = s1;
        a2[qi * DIM + h * 5 + d] = s2;
    }
    __syncthreads();

    // (x @ Wl3^T + bl3) summed over rows
    for (int d = tid; d < DIM; d += 256) {
        float s1 = 0.0f, s2 = 0.0f;
        for (int k = 0; k < KTOP; ++k) {
            float y1 = bl[3 * DIM + d], y2 = bl[3 * DIM + d];
            for (int c = 0; c < DIM; ++c) {
                y1 += a1[k * DIM + c] * Wl[(3 * DIM + d) * DIM + c];
                y2 += a2[k * DIM + c] * Wl[(3 * DIM + d) * DIM + c];
            }
            s1 += y1; s2 += y2;
        }
        r1[d] = s1; r2[d] = s2;
    }
    __syncthreads();

    if (tid < 2) {
        float s = bint[tid];
        for (int c = 0; c < DIM; ++c)
            s += r1[c] * Wint[tid * 2 * DIM + c] + r2[c] * Wint[tid * 2 * DIM + DIM + c];
        out[tid] = s;
    }
}

// ---------------------------------------------------------------------------
// Host orchestration
// ---------------------------------------------------------------------------
extern "C" void kernel_launch(void* const* d_in, const int* in_sizes, int n_in,
                              void* d_out, int out_size, void* d_ws, size_t ws_size,
                              hipStream_t stream) {
    const int*   p1     = (const int*)d_in[0];
    const int*   p2     = (const int*)d_in[1];
    const float* embed  = (const float*)d_in[2];
    const float* w_conv = (const float*)d_in[3];
    const float* b_conv = (const float*)d_in[4];
    const float* W_ff1  = (const float*)d_in[5];
    const float* b_ff1  = (const float*)d_in[6];
    const float* W_ff2  = (const float*)d_in[7];
    const float* b_ff2  = (const float*)d_in[8];
    const float* Wl     = (const float*)d_in[9];
    const float* bl     = (const float*)d_in[10];
    const float* W_int  = (const float*)d_in[11];
    const float* b_int  = (const float*)d_in[12];
    float* out = (float*)d_out;

    const int L = in_sizes[0];
    int nCh = (L + CHUNK - 1) / CHUNK;
    if (nCh > MAXCH) nCh = MAXCH;  // Stage-B LDS capacity (L=500k -> 62)

    // workspace carve-out (256B aligned)
    char* ws = (char*)d_ws;
    size_t off = 0;
    auto carve = [&](size_t bytes) {
        char* p = ws + off;
        off = (off + bytes + 255) & ~(size_t)255;
        return p;
    };
    float*    bufA  = (float*)carve((size_t)L * DIM * sizeof(float));
    float*    bufB  = (float*)carve((size_t)L * DIM * sizeof(float));
    float*    wfrag = (float*)carve(3 * KS * 3 * 64 * sizeof(float));
    unsigned* cand  = (unsigned*)carve((size_t)DIM * nCh * KTOP * sizeof(unsigned));
    float*    t1    = (float*)carve(KTOP * DIM * sizeof(float));
    float*    t2    = (float*)carve(KTOP * DIM * sizeof(float));
    (void)ws_size;

    prep_wfrag_kernel<<<1, 256, 0, stream>>>(w_conv, wfrag);

    const int convBlocks = (L + ROWS_BLK - 1) / ROWS_BLK;
    const size_t fragStride = (size_t)KS * 3 * 64;

    for (int seq = 0; seq < 2; ++seq) {
        const int* p = (seq == 0) ? p1 : p2;
        float* tout  = (seq == 0) ? t1 : t2;

        // layer 0 (fused gather), layer 1, layer 2
        conv_wmma_kernel<<<convBlocks, 256, 0, stream>>>(
            bufA, p, embed, wfrag + 0 * fragStride, b_conv, 0, bufA, L, 1);
        conv_wmma_kernel<<<convBlocks, 256, 0, stream>>>(
            bufA, p, embed, wfrag + 1 * fragStride, b_conv, 1, bufB, L, 0);
        conv_wmma_kernel<<<convBlocks, 256, 0, stream>>>(
            bufB, p, embed, wfrag + 2 * fragStride, b_conv, 2, bufA, L, 0);

        // top-k per column
        dim3 gA(nCh, DIM);
        topk_stageA_kernel<<<gA, 256, 0, stream>>>(bufA, cand, L, nCh);
        topk_stageB_kernel<<<DIM, 256, 0, stream>>>(cand, tout, nCh);
    }

    head_kernel<<<1, 256, 0, stream>>>(t1, t2, W_ff1, b_ff1, W_ff2, b_ff2,
                                       Wl, bl, W_int, b_int, out);
}